// RecsysContinuousPromptModel_68710886801851
// MI455X (gfx1250) — compile-verified
//
#include <hip/hip_runtime.h>
#include <hip/hip_bf16.h>
#include <stdint.h>

#define EMB   64
#define KHOPS 3
#define WPB   8   // waves per block in the gather-SpMM kernel

// ---------------- CDNA5 async global->LDS path (guarded) ----------------
#if __has_builtin(__builtin_amdgcn_global_load_async_to_lds_b64)
#define HAVE_ASYNC_LDS 1
#endif

// builtin signature (from hipcc diagnostic): (v2i AS1*, v2i AS3*, imm offset, imm cpol)
typedef int v2i __attribute__((vector_size(8)));
typedef __attribute__((address_space(1))) v2i GV2;  // global
typedef __attribute__((address_space(3))) v2i LV2;  // LDS

template <int N>
__device__ __forceinline__ void wait_async() {
#if __has_builtin(__builtin_amdgcn_s_wait_asynccnt)
  __builtin_amdgcn_s_wait_asynccnt(N);
#else
  asm volatile("s_wait_asynccnt %0" ::"i"(N) : "memory");
#endif
  asm volatile("" ::: "memory");  // keep LDS reads below the wait
}

#if defined(HAVE_ASYNC_LDS)
__device__ __forceinline__ void async_load_b64(const float2* src, float2* lds_dst) {
  __builtin_amdgcn_global_load_async_to_lds_b64((GV2*)src, (LV2*)lds_dst, 0, 0);
}
#endif

// ---------------- pipeline kernels ----------------

__global__ void zero_i32(int* __restrict__ p, int n) {
  int i = blockIdx.x * blockDim.x + threadIdx.x;
  if (i < n) p[i] = 0;
}

__global__ void count_rows(const int* __restrict__ erow, int* __restrict__ counts, int E) {
  int i = blockIdx.x * blockDim.x + threadIdx.x;
  if (i < E) atomicAdd(&counts[erow[i]], 1);
}

// Single-workgroup exclusive scan over N=150K counts (runs in microseconds).
__global__ void __launch_bounds__(1024) scan_excl(const int* __restrict__ counts,
                                                  int* __restrict__ row_start, int N) {
  __shared__ int sbuf[1024];
  __shared__ int carry_s;
  const int tid = threadIdx.x;
  if (tid == 0) carry_s = 0;
  __syncthreads();
  for (int base = 0; base < N; base += 1024) {
    int i = base + tid;
    int v = (i < N) ? counts[i] : 0;
    sbuf[tid] = v;
    __syncthreads();
    for (int off = 1; off < 1024; off <<= 1) {   // Hillis-Steele inclusive scan
      int t = (tid >= off) ? sbuf[tid - off] : 0;
      __syncthreads();
      sbuf[tid] += t;
      __syncthreads();
    }
    int carry = carry_s;
    if (i < N) row_start[i] = carry + sbuf[tid] - v;  // exclusive
    __syncthreads();
    if (tid == 0) carry_s = carry + sbuf[1023];
    __syncthreads();
  }
  if (tid == 0) row_start[N] = carry_s;
}

__global__ void dinv_kernel(const int* __restrict__ counts, float* __restrict__ dinv, int N) {
  int i = blockIdx.x * blockDim.x + threadIdx.x;
  if (i < N) dinv[i] = (counts[i] > 0) ? rsqrtf((float)counts[i]) : 0.0f;
}

// Scatter edges into CSR slots; pack (weight, col) as float2 so the hop kernel
// streams one 8-byte element per edge.
__global__ void fill_csr(const int* __restrict__ erow, const int* __restrict__ ecol,
                         const float* __restrict__ dinv, const int* __restrict__ row_start,
                         int* __restrict__ cursor, float2* __restrict__ csr, int E) {
  int i = blockIdx.x * blockDim.x + threadIdx.x;
  if (i >= E) return;
  int r = erow[i], c = ecol[i];
  float w = dinv[r] * dinv[c];
  int pos = atomicAdd(&cursor[r], 1);
  float2 p;
  p.x = w;
  p.y = __int_as_float(c);
  csr[row_start[r] + pos] = p;
}

__global__ void init_emb(const float* __restrict__ u, const float* __restrict__ it,
                         float* __restrict__ emb, int nu_elems, int total) {
  int i = blockIdx.x * blockDim.x + threadIdx.x;
  if (i < total) emb[i] = (i < nu_elems) ? u[i] : it[i - nu_elems];
}

// One wave32 per destination row; lane l owns dims {2l, 2l+1} (float2).
// Edge (w, col) pairs are double-buffered into LDS with async loads and
// broadcast across the wave with __shfl. Destination written exactly once;
// no global atomics in the hot loop, everything L2-resident.
__global__ void __launch_bounds__(WPB * 32)
spmm_gather(const int* __restrict__ row_start, const float2* __restrict__ csr,
            const float2* __restrict__ embc, float2* __restrict__ embn, int N) {
  __shared__ float2 abuf[WPB][2][32];
  const int wv   = threadIdx.x >> 5;
  const int lane = threadIdx.x & 31;
  const int rowi = blockIdx.x * WPB + wv;
  if (rowi >= N) return;
  const int s = row_start[rowi];
  const int e = row_start[rowi + 1];
  float2 a = make_float2(0.f, 0.f);

#if defined(HAVE_ASYNC_LDS)
  if (s < e) {
    {  // preload first chunk (clamped; tail lanes duplicate last edge, unused below)
      int idx = s + lane;
      if (idx > e - 1) idx = e - 1;
      async_load_b64(&csr[idx], &abuf[wv][0][lane]);
    }
    int cur = 0;
    for (int base = s; base < e; base += 32) {
      int nb = base + 32;
      if (nb < e) {  // kick next chunk, then wait for the older one (in-order completion)
        int idx = nb + lane;
        if (idx > e - 1) idx = e - 1;
        async_load_b64(&csr[idx], &abuf[wv][cur ^ 1][lane]);
        wait_async<1>();
      } else {
        wait_async<0>();
      }
      float2 p = abuf[wv][cur][lane];
      int n = e - base;
      if (n > 32) n = 32;
      for (int j = 0; j < n; ++j) {
        float w = __shfl(p.x, j, 32);
        int   c = __shfl(__float_as_int(p.y), j, 32);
        float2 v = embc[c * 32 + lane];   // coalesced 256B row read, L2-resident
        a.x = fmaf(w, v.x, a.x);
        a.y = fmaf(w, v.y, a.y);
      }
      cur ^= 1;
    }
  }
#else
  for (int base = s; base < e; base += 32) {
    int idx = base + lane;
    float2 p = (idx < e) ? csr[idx] : make_float2(0.f, __int_as_float(0));
    int n = e - base;
    if (n > 32) n = 32;
    for (int j = 0; j < n; ++j) {
      float w = __shfl(p.x, j, 32);
      int   c = __shfl(__float_as_int(p.y), j, 32);
      float2 v = embc[c * 32 + lane];
      a.x = fmaf(w, v.x, a.x);
      a.y = fmaf(w, v.y, a.y);
    }
  }
#endif

  embn[rowi * 32 + lane] = a;
}

// Gather + layer-mean accumulate directly into d_out: out (+)= 0.25 * emb[rows].
// Replaces the full-N acc buffer (saves ~230 MB of L2 traffic across hops).
__global__ void gather_accum(const float* __restrict__ emb, const int* __restrict__ uid,
                             const int* __restrict__ iid, float* __restrict__ out,
                             int B, int num_users, int add) {
  int i = blockIdx.x * blockDim.x + threadIdx.x;
  if (i >= B * 2 * EMB) return;
  int b = i >> 7, d = i & 127;
  int rowi = (d < EMB) ? uid[b] : (num_users + iid[b]);
  int cc = d & (EMB - 1);
  float v = 0.25f * emb[rowi * EMB + cc];  // /(K_HOPS+1)
  out[i] = add ? (out[i] + v) : v;
}

// ---------------- host side ----------------
extern "C" void kernel_launch(void* const* d_in, const int* in_sizes, int n_in,
                              void* d_out, int out_size, void* d_ws, size_t ws_size,
                              hipStream_t stream) {
  const float* users = (const float*)d_in[0];
  const float* items = (const float*)d_in[1];
  const int*   erow  = (const int*)d_in[2];
  const int*   ecol  = (const int*)d_in[3];
  const int*   uid   = (const int*)d_in[4];
  const int*   iid   = (const int*)d_in[5];
  const int nu = in_sizes[0] / EMB;
  const int ni = in_sizes[1] / EMB;
  const int N  = nu + ni;
  const int E  = in_sizes[2];
  const int B  = in_sizes[4];
  (void)n_in; (void)out_size; (void)ws_size;

  // workspace carve-out (~110 MB; whole working set is L2-resident on MI455X)
  char* ws = (char*)d_ws;
  size_t off = 0;
  auto alloc = [&](size_t bytes) -> char* {
    char* p = ws + off;
    off += (bytes + 255) & ~size_t(255);
    return p;
  };
  float*  embA   = (float*)alloc((size_t)N * EMB * 4);
  float*  embB   = (float*)alloc((size_t)N * EMB * 4);
  float2* csr    = (float2*)alloc((size_t)E * 8);
  int*    counts = (int*)alloc((size_t)N * 4);
  int*    cursor = (int*)alloc((size_t)N * 4);
  int*    rstart = (int*)alloc((size_t)(N + 1) * 4);
  float*  dinv   = (float*)alloc((size_t)N * 4);

  auto cdiv = [](int a, int b) { return (a + b - 1) / b; };
  const int gridB = cdiv(B * 2 * EMB, 256);

  // 1) degree histogram (== deg from edge_row) + cursor reset
  zero_i32<<<cdiv(N, 256), 256, 0, stream>>>(counts, N);
  zero_i32<<<cdiv(N, 256), 256, 0, stream>>>(cursor, N);
  count_rows<<<cdiv(E, 256), 256, 0, stream>>>(erow, counts, E);

  // 2) row_start = exclusive_scan(counts)
  scan_excl<<<1, 1024, 0, stream>>>(counts, rstart, N);

  // 3) dinv = deg^{-1/2} (0 where deg==0)
  dinv_kernel<<<cdiv(N, 256), 256, 0, stream>>>(counts, dinv, N);

  // 4) CSR fill with fused edge weights
  fill_csr<<<cdiv(E, 256), 256, 0, stream>>>(erow, ecol, dinv, rstart, cursor, csr, E);

  // 5) emb0 = concat(users, items); out = 0.25*emb0[gather]
  const int total = N * EMB;
  init_emb<<<cdiv(total, 256), 256, 0, stream>>>(users, items, embA, nu * EMB, total);
  gather_accum<<<gridB, 256, 0, stream>>>(embA, uid, iid, (float*)d_out, B, nu, /*add=*/0);

  // 6) K hops of gather-SpMM (ping-pong); after each hop, out += 0.25*emb_k[gather]
  const int gridR = cdiv(N, WPB);
  spmm_gather<<<gridR, WPB * 32, 0, stream>>>(rstart, csr, (const float2*)embA,
                                              (float2*)embB, N);
  gather_accum<<<gridB, 256, 0, stream>>>(embB, uid, iid, (float*)d_out, B, nu, 1);

  spmm_gather<<<gridR, WPB * 32, 0, stream>>>(rstart, csr, (const float2*)embB,
                                              (float2*)embA, N);
  gather_accum<<<gridB, 256, 0, stream>>>(embA, uid, iid, (float*)d_out, B, nu, 1);

  spmm_gather<<<gridR, WPB * 32, 0, stream>>>(rstart, csr, (const float2*)embA,
                                              (float2*)embB, N);
  gather_accum<<<gridB, 256, 0, stream>>>(embB, uid, iid, (float*)d_out, B, nu, 1);
}